// BitLevelMapper_27668179321269
// MI455X (gfx1250) — compile-verified
//
#include <hip/hip_runtime.h>
#include <stdint.h>

// ---------------------------------------------------------------------------
// BitLevelMapper for MI455X (gfx1250).
// Memory-bound: 128MB in + 128MB out @ 23.3 TB/s ~= 11us floor. No matrix
// structure -> no WMMA. CDNA5 win: the live table footprint is the compacted
// prefixes tables[i][0 .. 2^i-1] = 65535 floats (~256KB), which fits the
// 320KB/WGP LDS. Stage it via the Tensor Data Mover (tensor_load_to_lds,
// TENSORcnt), then serve all 33.5M random 4B gathers from LDS.
// Compacted base of table i is (2^i - 1) == mask, so lds index = mask + addr.
// ---------------------------------------------------------------------------

typedef __attribute__((ext_vector_type(4))) unsigned int u32x4;
typedef __attribute__((ext_vector_type(8))) int          i32x8;
typedef __attribute__((ext_vector_type(4))) int          i32x4;

#define NBITS            16
#define TABLE_SIZE       32768      // 1 << (NBITS-1)
#define LUT_WORDS        65535      // sum_{i=0}^{15} 2^i  (~256KB of LDS)
#define THREADS          1024
#define ROWS_PER_THREAD  8
#define ROWS_PER_BLOCK   (THREADS * ROWS_PER_THREAD)   // 8192 -> 256 blocks for B=2^21

__global__ __launch_bounds__(THREADS, 1)
void BitLevelMapper_27668179321269_kernel(const int* __restrict__ bits,
                                          const float* __restrict__ tables,
                                          float* __restrict__ out,
                                          int B)
{
    __shared__ float lut[LUT_WORDS];   // 262,140 bytes of the 320KB/WGP LDS

    // ---------------- Stage compacted table prefixes into LDS ----------------
#if defined(__has_builtin) && __has_builtin(__builtin_amdgcn_tensor_load_to_lds)
    if (threadIdx.x < 32) {
        // One wave issues 16 TDM descriptors (EXEC ignored; one issue per wave).
        // D# layout per CDNA5 ISA.
        #pragma unroll
        for (int i = 0; i < NBITS; ++i) {
            const unsigned n        = 1u << i;                 // live prefix of table i
            const unsigned lds_byte = 4u * (n - 1u);           // lut[2^i - 1]
            const uint64_t gaddr    =
                (uint64_t)(uintptr_t)(tables + (size_t)i * TABLE_SIZE);

            u32x4 g0;
            g0[0] = 1u;                                        // count=1, user mode
            g0[1] = lds_byte;                                  // lds_addr [63:32]
            g0[2] = (unsigned)gaddr;                           // global_addr [95:64]
            g0[3] = ((unsigned)(gaddr >> 32) & 0x01FFFFFFu)    // global_addr [120:96]
                    | 0x80000000u;                             // type=2 ("image") [127:126]

            i32x8 g1;
            g1[0] = 0x20000;                                   // data_size=2 (4B), wg_mask=0
            g1[1] = (int)0x80000000;                           // tensor_dim0=32768 (lo16 @ [63:48])
            g1[2] = 0x00010000;                                // tensor_dim0 hi=0, tensor_dim1=1
            g1[3] = (int)(n << 16);                            // tile_dim0 = 2^i  @ [127:112]
            g1[4] = 1;                                         // tile_dim1 = 1
            g1[5] = TABLE_SIZE;                                // tensor_dim0_stride lo32
            g1[6] = 0;
            g1[7] = 0;

            i32x4 z4 = {0, 0, 0, 0};
#if __clang_major__ >= 23
            i32x8 z8 = {0, 0, 0, 0, 0, 0, 0, 0};               // amdgpu-toolchain: 6-arg form
            __builtin_amdgcn_tensor_load_to_lds(g0, g1, z4, z4, z8, 0);
#else
            __builtin_amdgcn_tensor_load_to_lds(g0, g1, z4, z4, 0);      // ROCm 7.2: 5-arg
#endif
        }
        __builtin_amdgcn_s_wait_tensorcnt(0);                  // s_wait_tensorcnt 0
        // Table 0's single live entry, written from VALU code. Correct (it IS
        // lut[0]) and, crucially, makes `lut` a *written* object so LLVM cannot
        // fold the dynamic ds_load gathers of TDM-written LDS to undef.
        if (threadIdx.x == 0) lut[0] = tables[0];
    }
    __asm__ __volatile__("" ::: "memory");                     // LDS written behind compiler's back
#else
    // Fallback: cooperative staging. idx+1 = 2^i + a  =>  i = log2(idx+1).
    for (int idx = (int)threadIdx.x; idx < LUT_WORDS; idx += THREADS) {
        unsigned v = (unsigned)idx + 1u;
        int      i = 31 - __clz((int)v);
        unsigned a = v - (1u << i);
        lut[idx] = tables[(size_t)i * TABLE_SIZE + a];
    }
#endif
    __syncthreads();

    // ------------------------------- Compute ---------------------------------
    const size_t base_row = (size_t)blockIdx.x * ROWS_PER_BLOCK + threadIdx.x;

    #pragma unroll
    for (int k = 0; k < ROWS_PER_THREAD; ++k) {
        const size_t row = base_row + (size_t)k * THREADS;
        if (row >= (size_t)B) continue;

        // 64B of bit columns per row via 4x global_load_b128
        const int4* bp = (const int4*)(bits + row * NBITS);
        int4 b0 = bp[0], b1 = bp[1], b2 = bp[2], b3 = bp[3];
        const int c[NBITS] = { b0.x, b0.y, b0.z, b0.w,
                               b1.x, b1.y, b1.z, b1.w,
                               b2.x, b2.y, b2.z, b2.w,
                               b3.x, b3.y, b3.z, b3.w };

        // MSB-first columns: column j holds bit (15-j)
        int value = 0;
        #pragma unroll
        for (int j = 0; j < NBITS; ++j) value |= (c[j] & 1) << (NBITS - 1 - j);

        float r[NBITS];
        #pragma unroll
        for (int i = 0; i < NBITS; ++i) {
            const int   mask = (1 << i) - 1;
            const float f    = lut[mask + (value & mask)];     // compacted base == mask
            const float ib   = (float)((value >> i) & 1);
            r[NBITS - 1 - i] = ib + f - 2.0f * ib * f;         // ib XOR flip-probability
        }

        float4* op = (float4*)(out + row * NBITS);
        op[0] = make_float4(r[0],  r[1],  r[2],  r[3]);
        op[1] = make_float4(r[4],  r[5],  r[6],  r[7]);
        op[2] = make_float4(r[8],  r[9],  r[10], r[11]);
        op[3] = make_float4(r[12], r[13], r[14], r[15]);
    }
}

extern "C" void kernel_launch(void* const* d_in, const int* in_sizes, int n_in,
                              void* d_out, int out_size, void* d_ws, size_t ws_size,
                              hipStream_t stream) {
    const int*   bits   = (const int*)d_in[0];     // (B, 16) int32, MSB-first
    const float* tables = (const float*)d_in[1];   // (16, 32768) float32
    float*       out    = (float*)d_out;           // (B, 16) float32

    const int B    = in_sizes[0] / NBITS;
    const int grid = (B + ROWS_PER_BLOCK - 1) / ROWS_PER_BLOCK;   // 256 for B=2^21

    BitLevelMapper_27668179321269_kernel<<<grid, THREADS, 0, stream>>>(bits, tables, out, B);
}